// GAT_24326694764623
// MI455X (gfx1250) — compile-verified
//
#include <hip/hip_runtime.h>
#include <hip/hip_bf16.h>

// ---------------------------------------------------------------------------
// GAT (3-layer, 4-head, HID=128) for MI455X / gfx1250 (wave32, WMMA).
// fp32 end-to-end; GEMMs use V_WMMA_F32_16X16X4_F32 (exact fp32 matrix path).
// Block = 8 waves; each wave owns 4 N-tiles so the LDS A-fragment is reused
// 4x per ds_load and each A tile is staged exactly once per M-strip.
// ---------------------------------------------------------------------------

typedef __attribute__((ext_vector_type(2))) float v2f;
typedef __attribute__((ext_vector_type(8))) float v8f;

#define HID    128
#define HEADS  4
#define EMB    32
#define IN_DIM 16
#define NEG_SLOPE 0.2f
#define EPS_BN 1e-5f

// Monotone float -> uint key so unsigned atomicMax == float max.
__device__ __forceinline__ unsigned f2key(float f) {
    unsigned u = __float_as_uint(f);
    return (u & 0x80000000u) ? ~u : (u | 0x80000000u);
}
__device__ __forceinline__ float key2f(unsigned k) {
    return (k & 0x80000000u) ? __uint_as_float(k & 0x7FFFFFFFu)
                             : __uint_as_float(~k);
}

__device__ __forceinline__ void edge_sd(const int* __restrict__ ei, int e, int E,
                                        int& s, int& d) {
    if (e < E) { s = ei[e]; d = ei[E + e]; }
    else       { s = d = e - E; }            // self loops appended
}

// ---------------------------------------------------------------------------
// h0 = relu([emb, x@Wf.T + bf] @ Wc.T + bc)   — one 128-thread block per node
// ---------------------------------------------------------------------------
__global__ void __launch_bounds__(128)
k_encode(const float* __restrict__ x, const float* __restrict__ emb,
         const float* __restrict__ Wf, const float* __restrict__ bf,
         const float* __restrict__ Wc, const float* __restrict__ bc,
         float* __restrict__ h, int N)
{
    int n = blockIdx.x;
    if (n >= N) return;
    int t = threadIdx.x;
    __shared__ float cat[2 * EMB];           // [emb(32) | feat(32)]
    if (t < EMB) {                           // feat = x @ Wf.T + bf
        float acc = bf[t];
        #pragma unroll
        for (int j = 0; j < IN_DIM; ++j)
            acc += x[(size_t)n * IN_DIM + j] * Wf[t * IN_DIM + j];
        cat[EMB + t] = acc;
    } else if (t < 2 * EMB) {
        cat[t - EMB] = emb[(size_t)n * EMB + (t - EMB)];
    }
    __syncthreads();
    float acc = bc[t];
    #pragma unroll
    for (int j = 0; j < 2 * EMB; ++j)
        acc += cat[j] * Wc[t * (2 * EMB) + j];
    h[(size_t)n * HID + t] = fmaxf(acc, 0.0f);
}

// ---------------------------------------------------------------------------
// hp = h @ W.T   (N x 128) x (128 x 512) via v_wmma_f32_16x16x4_f32.
// Block = 256 thr = 8 waves, one block per 16-row M-strip covering ALL 512
// output columns: wave w owns N-tiles {w, w+8, w+16, w+24}. The 16x128 fp32
// A tile is staged once in LDS (row stride 132 to dodge bank conflicts) and
// each A fragment ds_load feeds 4 WMMAs (128 WMMAs per wave total).
//
// Fragment layouts per CDNA5 ISA 7.12.2:
//   A 16x4 : lane r(0..15) v0=A[r][k+0] v1=A[r][k+1]; lanes 16..31 -> k+2,k+3
//   B 4x16 : lane n(0..15) v0=B[k+0][n] v1=B[k+1][n]; lanes 16..31 -> k+2,k+3
//            (B[k][n] == W[n][k], W row-major (512,128))
//   D 16x16: acc[j] = D[m0 + j + 8*hi][n0 + (lane&15)]
// ---------------------------------------------------------------------------
__global__ void __launch_bounds__(256)
k_gemm_hp(const float* __restrict__ h, const float* __restrict__ W,
          float* __restrict__ hp, int N)
{
    __shared__ float As[16 * 132];
    const int tid = threadIdx.x;
    const int m0  = blockIdx.x * 16;

    // Stage the 16x128 A tile (512 float4 slots, 2 per thread).
    #pragma unroll
    for (int i = 0; i < 2; ++i) {
        int idx = tid + i * 256;             // 0..511
        int r   = idx >> 5;                  // row 0..15
        int kq  = idx & 31;                  // float4 column 0..31
        int row = m0 + r; if (row >= N) row = N - 1;
        float4 v = reinterpret_cast<const float4*>(h + (size_t)row * HID)[kq];
        float* dp = &As[r * 132 + kq * 4];
        dp[0] = v.x; dp[1] = v.y; dp[2] = v.z; dp[3] = v.w;
    }
    __syncthreads();

    const int wave = tid >> 5;
    const int lane = tid & 31;
    const int nl   = lane & 15;              // M-row (A) / N-col (B)
    const int hi   = lane >> 4;              // lane-half -> K offset 2*hi

    // 4 column tiles per wave: n0(t) = (wave + 8*t) * 16
    const float* B0 = W + (size_t)((wave      ) * 16 + nl) * HID;
    const float* B1 = W + (size_t)((wave +  8 ) * 16 + nl) * HID;
    const float* B2 = W + (size_t)((wave + 16 ) * 16 + nl) * HID;
    const float* B3 = W + (size_t)((wave + 24 ) * 16 + nl) * HID;

    v8f acc0 = {0.f,0.f,0.f,0.f,0.f,0.f,0.f,0.f};
    v8f acc1 = acc0, acc2 = acc0, acc3 = acc0;

    #pragma unroll
    for (int kk = 0; kk < HID; kk += 4) {
        const int kb = kk + 2 * hi;
        v2f a  = *reinterpret_cast<const v2f*>(&As[nl * 132 + kb]);
        v2f b0 = *reinterpret_cast<const v2f*>(B0 + kb);
        v2f b1 = *reinterpret_cast<const v2f*>(B1 + kb);
        v2f b2 = *reinterpret_cast<const v2f*>(B2 + kb);
        v2f b3 = *reinterpret_cast<const v2f*>(B3 + kb);
        acc0 = __builtin_amdgcn_wmma_f32_16x16x4_f32(false, a, false, b0,
                                                     (short)0, acc0, false, false);
        acc1 = __builtin_amdgcn_wmma_f32_16x16x4_f32(false, a, false, b1,
                                                     (short)0, acc1, false, false);
        acc2 = __builtin_amdgcn_wmma_f32_16x16x4_f32(false, a, false, b2,
                                                     (short)0, acc2, false, false);
        acc3 = __builtin_amdgcn_wmma_f32_16x16x4_f32(false, a, false, b3,
                                                     (short)0, acc3, false, false);
    }

    #pragma unroll
    for (int t = 0; t < 4; ++t) {
        const v8f acc = (t == 0) ? acc0 : (t == 1) ? acc1 : (t == 2) ? acc2 : acc3;
        const int n0  = (wave + 8 * t) * 16;
        float* outp = hp + (size_t)(m0 + 8 * hi) * (HEADS * HID) + n0 + nl;
        #pragma unroll
        for (int j = 0; j < 8; ++j) {
            int m = m0 + 8 * hi + j;
            if (m < N)
                outp[(size_t)j * (HEADS * HID)] = acc[j];
        }
    }
}

// ---------------------------------------------------------------------------
// al_s[n,h] = sum_c hp[n,h,c]*a_src[h,c]; al_d likewise. Warp per (node,head).
// ---------------------------------------------------------------------------
__global__ void __launch_bounds__(128)
k_attn_logits(const float* __restrict__ hp, const float* __restrict__ a_s,
              const float* __restrict__ a_d, float* __restrict__ als,
              float* __restrict__ ald, int N)
{
    int n = blockIdx.x;
    if (n >= N) return;
    int t = threadIdx.x, hh = t >> 5, lane = t & 31;
    const float* row = hp + (size_t)n * (HEADS * HID) + hh * HID;
    float ps = 0.f, pd = 0.f;
    #pragma unroll
    for (int i = 0; i < 4; ++i) {
        float v = row[lane + i * 32];
        ps += v * a_s[hh * HID + lane + i * 32];
        pd += v * a_d[hh * HID + lane + i * 32];
    }
    #pragma unroll
    for (int off = 16; off; off >>= 1) {
        ps += __shfl_xor(ps, off, 32);
        pd += __shfl_xor(pd, off, 32);
    }
    if (lane == 0) { als[n * HEADS + hh] = ps; ald[n * HEADS + hh] = pd; }
}

// e = leaky_relu(al_s[src]+al_d[dst]); store e; segment max via uint atomicMax
__global__ void __launch_bounds__(256)
k_edge_max(const int* __restrict__ ei, const float* __restrict__ als,
           const float* __restrict__ ald, unsigned* __restrict__ mkey,
           float* __restrict__ ebuf, int E, int T)
{
    int e = blockIdx.x * blockDim.x + threadIdx.x;
    if (e >= T) return;
    int s, d; edge_sd(ei, e, E, s, d);
    #pragma unroll
    for (int hh = 0; hh < HEADS; ++hh) {
        float v = als[s * HEADS + hh] + ald[d * HEADS + hh];
        v = (v > 0.f) ? v : NEG_SLOPE * v;
        ebuf[(size_t)e * HEADS + hh] = v;
        atomicMax(&mkey[d * HEADS + hh], f2key(v));
    }
}

// ex = exp(e - m[dst]); store ex; denom[dst] += ex
__global__ void __launch_bounds__(256)
k_edge_exp(const int* __restrict__ ei, float* __restrict__ ebuf,
           const unsigned* __restrict__ mkey, float* __restrict__ dnm,
           int E, int T)
{
    int e = blockIdx.x * blockDim.x + threadIdx.x;
    if (e >= T) return;
    int s, d; edge_sd(ei, e, E, s, d);
    (void)s;
    #pragma unroll
    for (int hh = 0; hh < HEADS; ++hh) {
        float m  = key2f(mkey[d * HEADS + hh]);
        float ex = expf(ebuf[(size_t)e * HEADS + hh] - m);
        ebuf[(size_t)e * HEADS + hh] = ex;
        atomicAdd(&dnm[d * HEADS + hh], ex);
    }
}

// alpha = ex / (denom[dst] + 1e-16)
__global__ void __launch_bounds__(256)
k_edge_alpha(const int* __restrict__ ei, float* __restrict__ ebuf,
             const float* __restrict__ dnm, int E, int T)
{
    int idx = blockIdx.x * blockDim.x + threadIdx.x;
    if (idx >= T * HEADS) return;
    int e = idx >> 2, hh = idx & 3;
    int s, d; edge_sd(ei, e, E, s, d);
    (void)s;
    ebuf[idx] /= (dnm[d * HEADS + hh] + 1e-16f);
}

// agg[dst,c] += (1/H) * sum_h alpha[e,h] * hp[src,h,c]   (thread per (e,c))
__global__ void __launch_bounds__(256)
k_edge_agg(const int* __restrict__ ei, const float* __restrict__ ebuf,
           const float* __restrict__ hp, float* __restrict__ agg,
           int E, int T)
{
    long long idx = (long long)blockIdx.x * blockDim.x + threadIdx.x;
    if (idx >= (long long)T * HID) return;
    int c = (int)(idx & (HID - 1));
    int e = (int)(idx >> 7);
    int s, d; edge_sd(ei, e, E, s, d);
    const float* hrow = hp + (size_t)s * (HEADS * HID) + c;
    const float* al   = ebuf + (size_t)e * HEADS;
    float v = al[0] * hrow[0]       + al[1] * hrow[HID]
            + al[2] * hrow[2 * HID] + al[3] * hrow[3 * HID];
    atomicAdd(&agg[(size_t)d * HID + c], 0.25f * v);
}

// per-channel sum / sumsq of (agg + bl) — thread = channel, block = node chunk
__global__ void __launch_bounds__(128)
k_bn_stats(const float* __restrict__ agg, const float* __restrict__ bl,
           float* __restrict__ sums, int N)
{
    int c  = threadIdx.x;
    int n0 = blockIdx.x * 256;
    int n1 = min(n0 + 256, N);
    float b = bl[c], s1 = 0.f, s2 = 0.f;
    for (int n = n0; n < n1; ++n) {
        float g = agg[(size_t)n * HID + c] + b;
        s1 += g; s2 += g * g;
    }
    atomicAdd(&sums[c], s1);
    atomicAdd(&sums[HID + c], s2);
}

// hnew = relu((g-mu)*rsqrt(var+eps)*gamma + beta) + hprev
__global__ void __launch_bounds__(256)
k_bn_apply(const float* __restrict__ agg, const float* __restrict__ bl,
           const float* __restrict__ gamma, const float* __restrict__ beta,
           const float* __restrict__ sums, const float* __restrict__ hprev,
           float* __restrict__ hnew, int N)
{
    long long idx = (long long)blockIdx.x * blockDim.x + threadIdx.x;
    if (idx >= (long long)N * HID) return;
    int c = (int)(idx & (HID - 1));
    float invN = 1.0f / (float)N;
    float mu   = sums[c] * invN;
    float var  = sums[HID + c] * invN - mu * mu;
    float g    = agg[idx] + bl[c];
    float y    = (g - mu) * rsqrtf(var + EPS_BN) * gamma[c] + beta[c];
    hnew[idx]  = fmaxf(y, 0.0f) + hprev[idx];
}

// out[n] = clip(h[n,:] . Wout + bout, -10, 10) — warp per node
__global__ void __launch_bounds__(256)
k_out(const float* __restrict__ h, const float* __restrict__ Wout,
      const float* __restrict__ bout, float* __restrict__ out, int N)
{
    int warp = (int)((blockIdx.x * (long long)blockDim.x + threadIdx.x) >> 5);
    int lane = threadIdx.x & 31;
    if (warp >= N) return;
    const float* row = h + (size_t)warp * HID;
    float p = 0.f;
    #pragma unroll
    for (int i = 0; i < 4; ++i)
        p += row[lane + i * 32] * Wout[lane + i * 32];
    #pragma unroll
    for (int off = 16; off; off >>= 1)
        p += __shfl_xor(p, off, 32);
    if (lane == 0) {
        float v = p + bout[0];
        out[warp] = fminf(fmaxf(v, -10.0f), 10.0f);
    }
}

// ---------------------------------------------------------------------------
extern "C" void kernel_launch(void* const* d_in, const int* in_sizes, int n_in,
                              void* d_out, int out_size, void* d_ws, size_t ws_size,
                              hipStream_t stream)
{
    const float* x     = (const float*)d_in[0];
    const int*   ei    = (const int*)  d_in[1];
    const float* emb   = (const float*)d_in[2];
    const float* Wf    = (const float*)d_in[3];
    const float* bf    = (const float*)d_in[4];
    const float* Wc    = (const float*)d_in[5];
    const float* bc    = (const float*)d_in[6];
    const float* Wl    = (const float*)d_in[7];
    const float* a_src = (const float*)d_in[8];
    const float* a_dst = (const float*)d_in[9];
    const float* bl    = (const float*)d_in[10];
    const float* gamma = (const float*)d_in[11];
    const float* beta  = (const float*)d_in[12];
    const float* Wout  = (const float*)d_in[13];
    const float* bout  = (const float*)d_in[14];
    float* out = (float*)d_out;

    const int N = in_sizes[0] / IN_DIM;     // 50000
    const int E = in_sizes[1] / 2;          // 400000
    const int T = E + N;                    // edges + self loops

    // Carve workspace (256B-aligned regions).
    char* p = (char*)d_ws;
    auto carve = [&](size_t bytes) -> void* {
        void* r = (void*)p;
        p += (bytes + 255) & ~(size_t)255;
        return r;
    };
    float*    h    = (float*)   carve((size_t)N * HID * 4);
    float*    h2   = (float*)   carve((size_t)N * HID * 4);
    float*    hp   = (float*)   carve((size_t)N * HEADS * HID * 4);
    float*    als  = (float*)   carve((size_t)N * HEADS * 4);
    float*    ald  = (float*)   carve((size_t)N * HEADS * 4);
    unsigned* mkey = (unsigned*)carve((size_t)N * HEADS * 4);
    float*    dnm  = (float*)   carve((size_t)N * HEADS * 4);
    float*    ebuf = (float*)   carve((size_t)T * HEADS * 4);
    float*    agg  = (float*)   carve((size_t)N * HID * 4);
    float*    sums = (float*)   carve((size_t)2 * HID * 4);
    (void)ws_size;

    k_encode<<<N, 128, 0, stream>>>(x, emb, Wf, bf, Wc, bc, h, N);

    float* cur = h;
    float* nxt = h2;
    const int mtiles = (N + 15) / 16;
    const int eg     = (T + 255) / 256;

    for (int l = 0; l < 3; ++l) {
        const float* Wlay = Wl + (size_t)l * (HEADS * HID) * HID;
        k_gemm_hp<<<mtiles, 256, 0, stream>>>(cur, Wlay, hp, N);
        k_attn_logits<<<N, 128, 0, stream>>>(hp, a_src + (size_t)l * HEADS * HID,
                                             a_dst + (size_t)l * HEADS * HID,
                                             als, ald, N);
        hipMemsetAsync(mkey, 0, (size_t)N * HEADS * 4, stream);
        hipMemsetAsync(dnm,  0, (size_t)N * HEADS * 4, stream);
        hipMemsetAsync(agg,  0, (size_t)N * HID * 4, stream);
        hipMemsetAsync(sums, 0, (size_t)2 * HID * 4, stream);

        k_edge_max  <<<eg, 256, 0, stream>>>(ei, als, ald, mkey, ebuf, E, T);
        k_edge_exp  <<<eg, 256, 0, stream>>>(ei, ebuf, mkey, dnm, E, T);
        k_edge_alpha<<<(T * HEADS + 255) / 256, 256, 0, stream>>>(ei, ebuf, dnm, E, T);
        k_edge_agg  <<<(int)(((long long)T * HID + 255) / 256), 256, 0, stream>>>(
            ei, ebuf, hp, agg, E, T);

        k_bn_stats<<<(N + 255) / 256, 128, 0, stream>>>(agg, bl + l * HID, sums, N);
        k_bn_apply<<<(int)(((long long)N * HID + 255) / 256), 256, 0, stream>>>(
            agg, bl + l * HID, gamma + l * HID, beta + l * HID, sums, cur, nxt, N);

        float* t2 = cur; cur = nxt; nxt = t2;
    }

    k_out<<<(N * 32 + 255) / 256, 256, 0, stream>>>(cur, Wout, bout, out, N);
    (void)n_in; (void)out_size;
}